// AGNNConv_68951404970011
// MI455X (gfx1250) — compile-verified
//
#include <hip/hip_runtime.h>
#include <math.h>

typedef __attribute__((ext_vector_type(2))) float v2f;
typedef __attribute__((ext_vector_type(8))) float v8f;

#define D_FEAT 32

// ---------------------------------------------------------------------------
// Pass 0: zero the output accumulator and the per-node softmax-denominator.
// ---------------------------------------------------------------------------
__global__ void agnn_init_zero(float* __restrict__ out, int n_out,
                               float* __restrict__ ssum, int n_s) {
    int i = blockIdx.x * blockDim.x + threadIdx.x;
    if (i < n_out) out[i] = 0.0f;
    if (i < n_s)   ssum[i] = 0.0f;
}

// ---------------------------------------------------------------------------
// Pass 1: L2-normalize node features. 8 lanes per node, float4 per lane,
// cross-lane reduction via shfl_xor within the 8-lane subgroup (wave32).
// ---------------------------------------------------------------------------
__global__ void agnn_l2norm(const float* __restrict__ feat,
                            float* __restrict__ normh, int N) {
    long t = (long)blockIdx.x * blockDim.x + threadIdx.x;
    long node = t >> 3;
    int  sub  = (int)(t & 7);
    if (node >= N) return;
    const float4 v = *(const float4*)(feat + node * D_FEAT + sub * 4);
    float ss = v.x * v.x + v.y * v.y + v.z * v.z + v.w * v.w;
    ss += __shfl_xor(ss, 1, 32);
    ss += __shfl_xor(ss, 2, 32);
    ss += __shfl_xor(ss, 4, 32);
    float nrm = fmaxf(sqrtf(ss), 1e-12f);   // torch F.normalize clamp
    float inv = 1.0f / nrm;
    float4 o = make_float4(v.x * inv, v.y * inv, v.z * inv, v.w * inv);
    *(float4*)(normh + node * D_FEAT + sub * 4) = o;
}

// ---------------------------------------------------------------------------
// Pass 2: per-edge cosine via WMMA. One wave32 handles a tile of 16 edges.
// Gram tile C = A(16x32) * B(32x16) built from 8 chained
// V_WMMA_F32_16X16X4_F32 ops (fp32, K=4 each); diag(C) = the 16 edge dots.
//
// fp32 16x4 A layout (ISA 7.12.2): lanes 0-15 hold K=0(v0),K=1(v1);
// lanes 16-31 hold K=2(v0),K=3(v1). B (4x16) mirrors with N=lane&15.
// So every lane loads a float2 at feature offset kb + ((lane>=16)?2:0)
// from row src[base+(lane&15)] (A) and row dst[base+(lane&15)] (B).
//
// C/D layout: vgpr r, lanes 0-15 -> (M=r, N=lane); lanes 16-31 ->
// (M=r+8, N=lane-16). Hence diag m<8 lives in lane m vgpr m; diag m>=8
// lives in lane m+16 vgpr m-8. Lanes {0..7, 24..31} are the writers.
//
// Then p = exp(beta*cos) is stored and atomically accumulated into
// ssum[dst]. (|beta*cos| <= 1 here, so the segment-max shift of the
// reference is a numerical no-op and the softmax is unchanged.)
// ---------------------------------------------------------------------------
__global__ void agnn_edge_softmax_num(const float* __restrict__ normh,
                                      const int*   __restrict__ src,
                                      const int*   __restrict__ dst,
                                      const float* __restrict__ beta,
                                      float* __restrict__ p,
                                      float* __restrict__ ssum,
                                      int E) {
    const int  lane = threadIdx.x & 31;
    const long wave = ((long)blockIdx.x * blockDim.x + threadIdx.x) >> 5;
    const long base = wave * 16;
    if (base >= E) return;                       // uniform per wave

    int em   = (int)base + (lane & 15);          // edge this half-lane maps to
    int eidx = (em < E) ? em : (E - 1);          // clamp for safe tail loads
    int srow = src[eidx];
    int drow = dst[eidx];

    const int koff = (lane >> 4) << 1;           // 0 for lanes 0-15, 2 for 16-31
    const float* arow = normh + (long)srow * D_FEAT + koff;
    const float* brow = normh + (long)drow * D_FEAT + koff;

    v8f c = {0.f, 0.f, 0.f, 0.f, 0.f, 0.f, 0.f, 0.f};
#pragma unroll
    for (int kb = 0; kb < D_FEAT; kb += 4) {
        v2f a = *(const v2f*)(arow + kb);
        v2f b = *(const v2f*)(brow + kb);
        // D = A*B + C : v_wmma_f32_16x16x4_f32
        c = __builtin_amdgcn_wmma_f32_16x16x4_f32(
                /*neg_a=*/false, a, /*neg_b=*/false, b,
                /*c_mod=*/(short)0, c, /*reuse_a=*/false, /*reuse_b=*/false);
    }

    // Extract this lane's diagonal element (unrolled cndmask select).
    int r = lane & 7;
    float dv = c[0];
#pragma unroll
    for (int i = 1; i < 8; ++i)
        if (r == i) dv = c[i];

    bool writer = (lane < 8) || (lane >= 24);
    if (writer && em < E) {
        float pe = __expf(beta[0] * dv);
        p[em] = pe;
        atomicAdd(&ssum[drow], pe);
    }
}

// ---------------------------------------------------------------------------
// Pass 3: weighted scatter-add. 8 lanes per edge, float4 per lane,
// w = p[e] / ssum[dst[e]] (ssum > 0: edge e itself contributed).
// ---------------------------------------------------------------------------
__global__ void agnn_aggregate(const float* __restrict__ feat,
                               const int*   __restrict__ src,
                               const int*   __restrict__ dst,
                               const float* __restrict__ p,
                               const float* __restrict__ ssum,
                               float* __restrict__ out, int E) {
    long t = (long)blockIdx.x * blockDim.x + threadIdx.x;
    long e = t >> 3;
    int  sub = (int)(t & 7);
    if (e >= E) return;
    int d = dst[e];
    float w = p[e] / ssum[d];
    const float4 f = *(const float4*)(feat + (long)src[e] * D_FEAT + sub * 4);
    float* o = out + (long)d * D_FEAT + sub * 4;
    atomicAdd(o + 0, f.x * w);
    atomicAdd(o + 1, f.y * w);
    atomicAdd(o + 2, f.z * w);
    atomicAdd(o + 3, f.w * w);
}

// ---------------------------------------------------------------------------
extern "C" void kernel_launch(void* const* d_in, const int* in_sizes, int n_in,
                              void* d_out, int out_size, void* d_ws, size_t ws_size,
                              hipStream_t stream) {
    const float* feat = (const float*)d_in[0];
    const float* beta = (const float*)d_in[1];
    const int*   src  = (const int*)d_in[2];
    const int*   dst  = (const int*)d_in[3];
    const int N = in_sizes[0] / D_FEAT;
    const int E = in_sizes[2];
    float* out = (float*)d_out;

    // Workspace: normh [N*32] | p [E] | ssum [N]   (~19.6 MB for the ref sizes)
    float* normh = (float*)d_ws;
    float* p     = normh + (size_t)N * D_FEAT;
    float* ssum  = p + (size_t)E;

    const int BLK = 256;

    // Pass 0: zero out + ssum (out_size >= N, one launch covers both).
    {
        int total = out_size > N ? out_size : N;
        int blocks = (total + BLK - 1) / BLK;
        agnn_init_zero<<<blocks, BLK, 0, stream>>>(out, out_size, ssum, N);
    }
    // Pass 1: normalize (8 threads / node).
    {
        long total = (long)N * 8;
        int blocks = (int)((total + BLK - 1) / BLK);
        agnn_l2norm<<<blocks, BLK, 0, stream>>>(feat, normh, N);
    }
    // Pass 2: WMMA cosine + exp + denominator (1 wave per 16 edges).
    {
        long tiles = ((long)E + 15) / 16;
        long waves_per_block = BLK / 32;
        int blocks = (int)((tiles + waves_per_block - 1) / waves_per_block);
        agnn_edge_softmax_num<<<blocks, BLK, 0, stream>>>(normh, src, dst, beta,
                                                          p, ssum, E);
    }
    // Pass 3: weighted scatter-add (8 threads / edge).
    {
        long total = (long)E * 8;
        int blocks = (int)((total + BLK - 1) / BLK);
        agnn_aggregate<<<blocks, BLK, 0, stream>>>(feat, src, dst, p, ssum, out, E);
    }
}